// SegmentedTensorSquareSelfInteraction_24902220383053
// MI455X (gfx1250) — compile-verified
//
#include <hip/hip_runtime.h>
#include <math.h>

// ---------------------------------------------------------------------------
// SegmentedTensorSquareSelfInteraction — fused fp32 WMMA kernel for gfx1250.
//
// Dead-code analysis of the reference: t1/t2/v1e/v2e/g1e/g2e are deleted
// before use, so only W2[:, :576] matters.  A block owns 16 nodes; 4 waves
// share one LDS buffer set (74 KB) and split the N-dimension of every GEMM,
// giving ~4 blocks/WGP (16 waves, 4 per SIMD32) for VMEM latency hiding.
// ---------------------------------------------------------------------------

typedef __attribute__((ext_vector_type(2))) float v2f;
typedef __attribute__((ext_vector_type(8))) float v8f;

#define MUL0     256
#define MUL1     128
#define GSEG     64
#define INV_DIM  448
#define NIRR     832
#define XDIM     640            // 256 + 3*128
#define EPSV     1e-6f

#define SAS  450                // s_all row stride (padded, even)
#define SHS  450                // h row stride (padded, even)
#define SOS  258                // o0/g1o row stride (padded, even)
#define VPS  130                // v1o / o1 plane row stride (padded, even)
#define BLOCK_LDS (16*SAS + 16*SHS + 16*SOS)   // 18528 floats = 74112 B
#define NWAVES 4                // waves per block, split the N dimension
#define ROWS   16               // nodes per block

static __device__ __forceinline__ v8f wmma4(v2f a, v2f b, v8f c) {
  // V_WMMA_F32_16X16X4_F32 : D = A(16x4) * B(4x16) + C(16x16), all f32
  return __builtin_amdgcn_wmma_f32_16x16x4_f32(
      /*neg_a=*/false, a, /*neg_b=*/false, b,
      /*c_mod=*/(short)0, c, /*reuse_a=*/false, /*reuse_b=*/false);
}

static __device__ __forceinline__ float wredsum(float v) {
  #pragma unroll
  for (int off = 16; off; off >>= 1) v += __shfl_xor(v, off, 32);
  return v;
}

// WGP-scope prefetch (scope 0): pulls into all cache levels on miss and
// stops on a WGP$ hit.  __builtin_prefetch can only express SE/SYS scopes,
// which skip the near cache, so use inline asm.
static __device__ __forceinline__ void prefetch_wgp(const void* p) {
  asm volatile("global_prefetch_b8 %0, off" :: "v"(p));
}

// One wave computes a 16x32 output tile: rows = block's 16 nodes, cols
// [n0, n0+32).  A streamed from LDS (row-major, stride lda), B streamed from
// global (row-major, stride ldb).  Two WMMAs per K-step share one A-fragment.
static __device__ __forceinline__ void gemm16x32(
    const float* Alds, int lda,
    const float* __restrict__ B, int ldb,
    int K, int n0, int lane, v8f& c0, v8f& c1)
{
  // Hoisted panel prefetch: lane l covers B row (i + l); each 32-col row
  // spans two 64-byte lines.  Keeps the WMMA loop free of extra VMEM ops.
  for (int i = 0; i < K; i += 32) {
    const char* pp = (const char*)(B + (long)(i + lane) * ldb + n0);
    prefetch_wgp(pp);
    prefetch_wgp(pp + 64);
  }

  const int rsel = lane & 15;          // A row / B col within tile
  const int kh   = (lane >> 4) << 1;   // lanes 16-31 carry K+2, K+3
  const float* pb = B + (long)kh * ldb + n0 + rsel;
  const float* pa = Alds + rsel * lda + kh;
  for (int k = 0; k < K; k += 4) {
    v2f a = *(const v2f*)(pa + k);
    v2f b0, b1;
    b0.x = pb[0];        b0.y = pb[ldb];
    b1.x = pb[16];       b1.y = pb[ldb + 16];
    pb += 4L * ldb;
    c0 = wmma4(a, b0, c0);
    c1 = wmma4(a, b1, c1);
  }
}

__global__ __launch_bounds__(NWAVES * 32)
void stssi_fused_kernel(const float* __restrict__ x,
                        const float* __restrict__ W1,
                        const float* __restrict__ W2,
                        const float* __restrict__ Wl0,
                        const float* __restrict__ Wl1,
                        float* __restrict__ out)
{
  const int lane = threadIdx.x & 31;
  const int wave = threadIdx.x >> 5;
  const int nodeBase = blockIdx.x * ROWS;

  __shared__ float smem[BLOCK_LDS];
  float* SA = smem;                 // [16][SAS]  s_all, later o1 stage
  float* SH = SA + 16 * SAS;        // [16][SHS]  h, later v1o planes
  float* SO = SH + 16 * SHS;        // [16][SOS]  g1o stage, then o0

  const float rIF = 1.0f / sqrtf((float)INV_DIM);   // 1/sqrt(448)
  const float rM1 = 1.0f / sqrtf((float)MUL1);      // 1/sqrt(128)
  const float ISQ3 = 0.57735026918962576f;
  const float SQ2  = 1.41421356237309505f;

  const int rowb = (lane >> 4) << 3;   // C-tile: VGPR j -> row rowb + j
  const int cl   = lane & 15;          // C-tile column within tile

  // ---- Phase 1: build s_all = [s, t0] into SA (rows split by wave) --------
  for (int r = wave; r < ROWS; r += NWAVES) {
    const float* xr = x + (long)(nodeBase + r) * XDIM;
    for (int c = lane; c < MUL0; c += 32)
      SA[r * SAS + c] = xr[c];
    for (int g = lane; g < GSEG; g += 32) {
      const float* pv = xr + MUL0 + 6 * g;
      float ax = pv[0], ay = pv[1], az = pv[2];
      float bx = pv[3], by = pv[4], bz = pv[5];
      float aa = ax*ax + ay*ay + az*az;
      float ab = ax*bx + ay*by + az*bz;
      float bb = bx*bx + by*by + bz*bz;
      SA[r * SAS + MUL0 + 3*g + 0] = aa * ISQ3;
      SA[r * SAS + MUL0 + 3*g + 1] = SQ2 * ab * ISQ3;
      SA[r * SAS + MUL0 + 3*g + 2] = bb * ISQ3;
    }
  }
  __syncthreads();

  // ---- Phase 2: h = silu(s_all @ W1 / sqrt(448)) --> SH -------------------
  for (int p = wave; p < INV_DIM / 32; p += NWAVES) {   // 14 tile-pairs
    const int n0 = 32 * p;
    v8f c0 = {}; v8f c1 = {};
    gemm16x32(SA, SAS, W1, INV_DIM, INV_DIM, n0, lane, c0, c1);
    #pragma unroll
    for (int j = 0; j < 8; ++j) {
      float h0 = c0[j] * rIF;
      h0 = h0 * __builtin_amdgcn_rcpf(1.0f + __expf(-h0));
      float h1 = c1[j] * rIF;
      h1 = h1 * __builtin_amdgcn_rcpf(1.0f + __expf(-h1));
      SH[(rowb + j) * SHS + n0 +      cl] = h0;
      SH[(rowb + j) * SHS + n0 + 16 + cl] = h1;
    }
  }
  __syncthreads();

  // ---- Phase 3: g0 = (h @ W2[:, :448]) / sqrt(448); s_all *= g0 -----------
  for (int p = wave; p < INV_DIM / 32; p += NWAVES) {   // 14 tile-pairs
    const int n0 = 32 * p;
    v8f c0 = {}; v8f c1 = {};
    gemm16x32(SH, SHS, W2, NIRR, INV_DIM, n0, lane, c0, c1);
    #pragma unroll
    for (int j = 0; j < 8; ++j) {
      SA[(rowb + j) * SAS + n0 +      cl] *= c0[j] * rIF;
      SA[(rowb + j) * SAS + n0 + 16 + cl] *= c1[j] * rIF;
    }
  }
  // ---- Phase 4: g1o = (h @ W2[:, 448:576]) / sqrt(448) --> SO stage -------
  for (int p = wave; p < MUL1 / 32; p += NWAVES) {      // 4 tile-pairs
    const int n0 = 32 * p;
    v8f c0 = {}; v8f c1 = {};
    gemm16x32(SH, SHS, W2 + INV_DIM, NIRR, INV_DIM, n0, lane, c0, c1);
    #pragma unroll
    for (int j = 0; j < 8; ++j) {
      SO[(rowb + j) * SOS + n0 +      cl] = c0[j] * rIF;
      SO[(rowb + j) * SOS + n0 + 16 + cl] = c1[j] * rIF;
    }
  }
  __syncthreads();   // all h reads done; g1o staged

  // ---- Phase 5: v1o planes (v * g1o) overwrite SH (h is dead) -------------
  for (int r = wave; r < ROWS; r += NWAVES) {
    const float* xr = x + (long)(nodeBase + r) * XDIM + MUL0;
    for (int m = lane; m < MUL1; m += 32) {
      float g = SO[r * SOS + m];
      SH[0 * 16 * VPS + r * VPS + m] = xr[3*m + 0] * g;
      SH[1 * 16 * VPS + r * VPS + m] = xr[3*m + 1] * g;
      SH[2 * 16 * VPS + r * VPS + m] = xr[3*m + 2] * g;
    }
  }
  __syncthreads();   // v1o staged; g1o consumed (SO reusable)

  // ---- Phase 6: o0 = (gated s_all @ Wl0) / sqrt(448) --> SO ---------------
  for (int p = wave; p < MUL0 / 32; p += NWAVES) {      // 8 tile-pairs
    const int n0 = 32 * p;
    v8f c0 = {}; v8f c1 = {};
    gemm16x32(SA, SAS, Wl0, MUL0, INV_DIM, n0, lane, c0, c1);
    #pragma unroll
    for (int j = 0; j < 8; ++j) {
      SO[(rowb + j) * SOS + n0 +      cl] = c0[j] * rIF;
      SO[(rowb + j) * SOS + n0 + 16 + cl] = c1[j] * rIF;
    }
  }
  __syncthreads();   // o0 complete; SA reads done (SA reusable)

  // ---- Phase 7: LayerNorm(o0) --> out[:, :256] ----------------------------
  for (int r = wave; r < ROWS; r += NWAVES) {
    float s = 0.f, ss = 0.f;
    for (int c = lane; c < MUL0; c += 32) {
      float v = SO[r * SOS + c];
      s += v;  ss += v * v;
    }
    s  = wredsum(s);
    ss = wredsum(ss);
    float mu  = s  * (1.0f / MUL0);
    float var = ss * (1.0f / MUL0) - mu * mu;
    float sc  = __builtin_amdgcn_rsqf(var + EPSV);
    float* o = out + (long)(nodeBase + r) * XDIM;
    for (int c = lane; c < MUL0; c += 32)
      o[c] = (SO[r * SOS + c] - mu) * sc;
  }

  // ---- Phase 8: o1_i = (v1o_i @ Wl1) / sqrt(128) --> SA (s_all is dead) ---
  for (int t = wave; t < 3 * (MUL1 / 32); t += NWAVES) { // 12 (plane, pair)
    const int i  = t >> 2;
    const int n0 = 32 * (t & 3);
    const float* Ai = SH + i * 16 * VPS;
    float*       Oi = SA + i * 16 * VPS;     // 3*2080 <= 7200 floats
    v8f c0 = {}; v8f c1 = {};
    gemm16x32(Ai, VPS, Wl1, MUL1, MUL1, n0, lane, c0, c1);
    #pragma unroll
    for (int j = 0; j < 8; ++j) {
      Oi[(rowb + j) * VPS + n0 +      cl] = c0[j] * rM1;
      Oi[(rowb + j) * VPS + n0 + 16 + cl] = c1[j] * rM1;
    }
  }
  __syncthreads();

  // ---- Phase 9: RMS norm over 384 --> out[:, 256:] ------------------------
  for (int r = wave; r < ROWS; r += NWAVES) {
    float ss = 0.f;
    for (int i = 0; i < 3; ++i)
      for (int m = lane; m < MUL1; m += 32) {
        float v = SA[i * 16 * VPS + r * VPS + m];
        ss += v * v;
      }
    ss = wredsum(ss);
    float inv = __builtin_amdgcn_rsqf(ss * (1.0f / (3 * MUL1)) + EPSV);
    float* o = out + (long)(nodeBase + r) * XDIM + MUL0;
    for (int m = lane; m < MUL1; m += 32) {
      o[3*m + 0] = SA[0 * 16 * VPS + r * VPS + m] * inv;
      o[3*m + 1] = SA[1 * 16 * VPS + r * VPS + m] * inv;
      o[3*m + 2] = SA[2 * 16 * VPS + r * VPS + m] * inv;
    }
  }
}

extern "C" void kernel_launch(void* const* d_in, const int* in_sizes, int n_in,
                              void* d_out, int out_size, void* d_ws, size_t ws_size,
                              hipStream_t stream) {
  const float* x   = (const float*)d_in[0];
  const float* W1  = (const float*)d_in[1];
  const float* W2  = (const float*)d_in[2];
  const float* Wl0 = (const float*)d_in[3];
  const float* Wl1 = (const float*)d_in[4];
  float* out = (float*)d_out;

  const int N = in_sizes[0] / XDIM;          // 32768
  dim3 grid((N + ROWS - 1) / ROWS);          // 2048 blocks, 16 nodes each
  dim3 block(NWAVES * 32);                   // 128 threads = 4 waves
  stssi_fused_kernel<<<grid, block, 0, stream>>>(x, W1, W2, Wl0, Wl1, out);
}